// MPNN4_10333691314706
// MI455X (gfx1250) — compile-verified
//
#include <hip/hip_runtime.h>

typedef __attribute__((ext_vector_type(2))) float v2f;
typedef __attribute__((ext_vector_type(8))) float v8f;

#define NFEAT 128
#define HID   64
#define NGRAPH 64

// ---------------------------------------------------------------------------
// init: agg = 0, deg = 1 (self-loop), u = 0
// ---------------------------------------------------------------------------
__global__ void k_init(float* __restrict__ agg, float* __restrict__ deg,
                       float* __restrict__ u, int nNodes) {
    long long total = (long long)nNodes * HID;
    long long stride = (long long)gridDim.x * blockDim.x;
    for (long long i = (long long)blockIdx.x * blockDim.x + threadIdx.x; i < total; i += stride)
        agg[i] = 0.0f;
    for (long long i = (long long)blockIdx.x * blockDim.x + threadIdx.x; i < nNodes; i += stride)
        deg[i] = 1.0f;
    for (long long i = (long long)blockIdx.x * blockDim.x + threadIdx.x; i < NGRAPH * HID; i += stride)
        u[i] = 0.0f;
}

// ---------------------------------------------------------------------------
// h = x @ W   via V_WMMA_F32_16X16X4_F32 (f32 in, f32 acc — matches reference)
// Block = 256 threads = 8 waves; each wave: 16 rows x 64 cols (4 N-tiles).
//
// W staged in LDS K-pair-interleaved:  Wl[kp][c] = {W[2kp][c], W[2kp+1][c]}
// so each B fragment (lane l needs {W[k+kh][n], W[k+kh+1][n]}, kh=(l>>4)*2)
// is ONE aligned 8-byte ds_load_b64 into an even VGPR pair — no packing movs.
//
// A 16x4 f32 layout: lane l -> M = l&15, K = (l>>4)*2 + {0,1}  (2 VGPRs)
// B  4x16 f32 layout: lane l -> N = l&15, K = (l>>4)*2 + {0,1}
// C/D 16x16 f32: VGPR r -> M = r + (lane>=16 ? 8:0), N = lane&15
// ---------------------------------------------------------------------------
__global__ __launch_bounds__(256) void k_gemm(const float* __restrict__ x,
                                              const float* __restrict__ W,
                                              float* __restrict__ h, int nNodes) {
    __shared__ float Wl[NFEAT * HID];   // 32 KB of the 320 KB/WGP pool
    // Interleaved fill: dest float2 entry idx -> (kp = idx/64, c = idx%64)
    for (int idx = threadIdx.x; idx < (NFEAT / 2) * HID; idx += 256) {
        int kp = idx >> 6;        // K pair index 0..63
        int c  = idx & 63;        // column 0..63
        v2f v = { W[(2 * kp) * HID + c], W[(2 * kp + 1) * HID + c] };
        *(v2f*)&Wl[(size_t)kp * (2 * HID) + 2 * c] = v;
    }
    __syncthreads();

    const int lane = threadIdx.x & 31;
    const int wave = threadIdx.x >> 5;
    const int m    = lane & 15;          // row within 16-tile (and B column)
    const int khh  = lane >> 4;          // K half: 0 or 1 (kh = 2*khh)

    const int row0 = blockIdx.x * 128 + wave * 16;
    int row  = row0 + m;
    int rowc = row < nNodes ? row : nNodes - 1;   // clamp (EXEC must stay all-1s for WMMA)
    const float* __restrict__ xr = x + (size_t)rowc * NFEAT;

    // per-lane invariant LDS base (in floats): row-pair select + column pair
    const float* __restrict__ wbase = &Wl[(size_t)khh * (2 * HID) + 2 * m];

    v8f acc0{}, acc1{}, acc2{}, acc3{};

    for (int k = 0; k < NFEAT; k += 4) {
        v2f a = *(const v2f*)(xr + k + 2 * khh);                 // 8B-aligned global
        const float* wk = wbase + (size_t)(k >> 1) * (2 * HID);  // this k-step's pair row

        v2f b0 = *(const v2f*)(wk + 0);    // tile 0: cols  0..15  (ds_load_b64)
        acc0 = __builtin_amdgcn_wmma_f32_16x16x4_f32(false, a, false, b0, (short)0, acc0, false, false);
        v2f b1 = *(const v2f*)(wk + 32);   // tile 1: cols 16..31
        acc1 = __builtin_amdgcn_wmma_f32_16x16x4_f32(false, a, false, b1, (short)0, acc1, false, false);
        v2f b2 = *(const v2f*)(wk + 64);   // tile 2: cols 32..47
        acc2 = __builtin_amdgcn_wmma_f32_16x16x4_f32(false, a, false, b2, (short)0, acc2, false, false);
        v2f b3 = *(const v2f*)(wk + 96);   // tile 3: cols 48..63
        acc3 = __builtin_amdgcn_wmma_f32_16x16x4_f32(false, a, false, b3, (short)0, acc3, false, false);
    }

    const int mhalf = khh * 8;
    #pragma unroll
    for (int r = 0; r < 8; ++r) {
        int rr = row0 + mhalf + r;
        if (rr < nNodes) {
            float* hr = h + (size_t)rr * HID + m;
            hr[0]  = acc0[r];
            hr[16] = acc1[r];
            hr[32] = acc2[r];
            hr[48] = acc3[r];
        }
    }
}

// ---------------------------------------------------------------------------
// deg[dst] += 1 per edge (self-loop pre-seeded to 1 by k_init)
// ---------------------------------------------------------------------------
__global__ void k_deg(const int* __restrict__ dst, float* __restrict__ deg, int nE) {
    int i = blockIdx.x * blockDim.x + threadIdx.x;
    if (i < nE) atomicAdd(&deg[dst[i]], 1.0f);
}

// dis = deg^-1/2 in place (deg >= 1 always)
__global__ void k_rsqrt(float* __restrict__ deg, int n) {
    int i = blockIdx.x * blockDim.x + threadIdx.x;
    if (i < n) deg[i] = rsqrtf(deg[i]);
}

// ---------------------------------------------------------------------------
// Edge scatter: agg[dst] += h[src] * dis[src]*dis[dst]
// 16 threads per edge, float4 per thread (64 feats). h/agg live in L2 (192MB).
// ---------------------------------------------------------------------------
__global__ void k_scatter(const int* __restrict__ src, const int* __restrict__ dst,
                          const float* __restrict__ h, const float* __restrict__ dis,
                          float* __restrict__ agg, int nE) {
    long long gid = (long long)blockIdx.x * blockDim.x + threadIdx.x;
    int e = (int)(gid >> 4);
    if (e >= nE) return;
    int q = (int)(gid & 15);
    int s = src[e], d = dst[e];
    float w = dis[s] * dis[d];
    float4 v = *(const float4*)(h + (size_t)s * HID + q * 4);
    float* ap = agg + (size_t)d * HID + q * 4;
    atomicAdd(ap + 0, v.x * w);
    atomicAdd(ap + 1, v.y * w);
    atomicAdd(ap + 2, v.z * w);
    atomicAdd(ap + 3, v.w * w);
}

// ---------------------------------------------------------------------------
// Node epilogue: hn = relu(agg + h*dis^2 + b); u[batch[node]] += hn (pooling)
// ---------------------------------------------------------------------------
__global__ void k_node(const float* __restrict__ agg, const float* __restrict__ h,
                       const float* __restrict__ dis, const float* __restrict__ b,
                       const int* __restrict__ batch, float* __restrict__ u, int nNodes) {
    long long i = (long long)blockIdx.x * blockDim.x + threadIdx.x;
    long long total = (long long)nNodes * HID;
    if (i >= total) return;
    int node = (int)(i >> 6);
    int j    = (int)(i & 63);
    float di = dis[node];
    float v = agg[i] + h[i] * di * di + b[j];
    v = v > 0.0f ? v : 0.0f;
    atomicAdd(&u[(size_t)batch[node] * HID + j], v);
}

// ---------------------------------------------------------------------------
// Tiny MLP head: out = relu(u @ W1 + b1) @ W2 + b2   (64 graphs, 65 KFLOP)
// ---------------------------------------------------------------------------
__global__ void k_mlp(const float* __restrict__ u, const float* __restrict__ W1,
                      const float* __restrict__ b1, const float* __restrict__ W2,
                      const float* __restrict__ b2, float* __restrict__ out) {
    int g = threadIdx.x;
    if (g >= NGRAPH) return;
    const float* ug = u + (size_t)g * HID;
    float o = b2[0];
    #pragma unroll
    for (int k = 0; k < 16; ++k) {
        float t = b1[k];
        for (int j = 0; j < HID; ++j)
            t += ug[j] * W1[j * 16 + k];
        t = t > 0.0f ? t : 0.0f;
        o += t * W2[k];
    }
    out[g] = o;
}

// ---------------------------------------------------------------------------
extern "C" void kernel_launch(void* const* d_in, const int* in_sizes, int n_in,
                              void* d_out, int out_size, void* d_ws, size_t ws_size,
                              hipStream_t stream) {
    const float* x   = (const float*)d_in[0];
    const float* W   = (const float*)d_in[1];
    const float* b   = (const float*)d_in[2];
    const float* W1  = (const float*)d_in[3];
    const float* b1  = (const float*)d_in[4];
    const float* W2  = (const float*)d_in[5];
    const float* b2  = (const float*)d_in[6];
    const int*   ei  = (const int*)d_in[7];
    const int*   bat = (const int*)d_in[8];

    const int nNodes = in_sizes[0] / NFEAT;
    const int nE     = in_sizes[7] / 2;
    const int* srcI = ei;
    const int* dstI = ei + nE;

    char* ws = (char*)d_ws;
    float* h   = (float*)ws;                                            // N*64 f32
    float* agg = (float*)(ws + (size_t)nNodes * HID * 4);               // N*64 f32
    float* deg = (float*)(ws + (size_t)nNodes * HID * 8);               // N f32 (deg -> dis in place)
    float* u   = (float*)(ws + (size_t)nNodes * HID * 8 + (size_t)nNodes * 4); // 64*64 f32

    k_init<<<1024, 256, 0, stream>>>(agg, deg, u, nNodes);

    int gemmBlocks = (nNodes + 127) / 128;
    k_gemm<<<gemmBlocks, 256, 0, stream>>>(x, W, h, nNodes);

    k_deg<<<(nE + 255) / 256, 256, 0, stream>>>(dstI, deg, nE);
    k_rsqrt<<<(nNodes + 255) / 256, 256, 0, stream>>>(deg, nNodes);

    long long scThreads = (long long)nE * 16;
    k_scatter<<<(int)((scThreads + 255) / 256), 256, 0, stream>>>(srcI, dstI, h, deg, agg, nE);

    long long ndThreads = (long long)nNodes * HID;
    k_node<<<(int)((ndThreads + 255) / 256), 256, 0, stream>>>(agg, h, deg, b, bat, u, nNodes);

    k_mlp<<<1, 64, 0, stream>>>(u, W1, b1, W2, b2, (float*)d_out);
}